// gumbelSoftMLP_57896159150713
// MI455X (gfx1250) — compile-verified
//
#include <hip/hip_runtime.h>
#include <math.h>

#define N_NODES 500
#define BATCH   512

typedef float f4 __attribute__((ext_vector_type(4)));

#if __has_builtin(__builtin_amdgcn_global_load_async_to_lds_b32)
#define HAVE_ASYNC_LDS 1
typedef __attribute__((address_space(1))) int gas_int;  // global AS pointee
typedef __attribute__((address_space(3))) int las_int;  // LDS AS pointee
#endif

// ---------------------------------------------------------------------------
// Main kernel (placed first so the disasm snippet shows the hot loop).
// grid = 512 (one WG per batch b), block = 128 (4 x wave32).
// Thread t < 125 owns columns i = 4t..4t+3. Streams u with NT b128 loads,
// prefetching 32 rows (64 KB) ahead every iteration; thr/Wt stay L2-resident.
// ---------------------------------------------------------------------------
__global__ __launch_bounds__(128) void gumbel_mlp_kernel(
    const float* __restrict__ x,
    const float* __restrict__ u,
    const float* __restrict__ thr,
    const float* __restrict__ Wt,
    float* __restrict__ out) {
  __shared__ float xs[N_NODES];

  const int b = blockIdx.x;
  const int t = threadIdx.x;
  const float* __restrict__ xrow = x + (size_t)b * N_NODES;

  // Stage x[b,:] into LDS via the CDNA5 async global->LDS DMA path.
  for (int idx = t; idx < N_NODES; idx += 128) {
#ifdef HAVE_ASYNC_LDS
    __builtin_amdgcn_global_load_async_to_lds_b32(
        (gas_int*)(xrow + idx),
        (las_int*)(&xs[idx]),
        0, 0);
#else
    xs[idx] = xrow[idx];
#endif
  }
#ifdef HAVE_ASYNC_LDS
#if __has_builtin(__builtin_amdgcn_s_wait_asynccnt)
  __builtin_amdgcn_s_wait_asynccnt(0);
#else
  asm volatile("s_wait_asynccnt 0" ::: "memory");
#endif
#endif
  __syncthreads();

  if (t < 125) {
    const int i0 = t * 4;
    const size_t ubase = (size_t)b * N_NODES * N_NODES + (size_t)i0;

    const float lo = 1e-6f;
    const float hi = 1.0f - 1e-6f;

    f4 acc = {0.0f, 0.0f, 0.0f, 0.0f};

#pragma unroll 8
    for (int j = 0; j < N_NODES; ++j) {
      // prefetch the u stream 32 rows ahead, EVERY row (contiguous stream)
      if (j + 32 < N_NODES) {
        __builtin_prefetch(u + ubase + (size_t)(j + 32) * N_NODES, 0, 0);
      }

      // single-use stream: non-temporal so L2 keeps thr/Wt resident
      f4 u4 = __builtin_nontemporal_load(
          (const f4*)(u + ubase + (size_t)j * N_NODES));
      f4 t4 = *(const f4*)(thr + j * N_NODES + i0);
      f4 w4 = *(const f4*)(Wt + j * N_NODES + i0);
      float xb = xs[j];

      float uc0 = fminf(fmaxf(u4.x, lo), hi);
      float uc1 = fminf(fmaxf(u4.y, lo), hi);
      float uc2 = fminf(fmaxf(u4.z, lo), hi);
      float uc3 = fminf(fmaxf(u4.w, lo), hi);

      // gate = hard gumbel-sigmoid sample AND off-diagonal mask (j != i)
      acc.x = fmaf(w4.x, ((uc0 > t4.x) && (j != i0 + 0)) ? xb : 0.0f, acc.x);
      acc.y = fmaf(w4.y, ((uc1 > t4.y) && (j != i0 + 1)) ? xb : 0.0f, acc.y);
      acc.z = fmaf(w4.z, ((uc2 > t4.z) && (j != i0 + 2)) ? xb : 0.0f, acc.z);
      acc.w = fmaf(w4.w, ((uc3 > t4.w) && (j != i0 + 3)) ? xb : 0.0f, acc.w);
    }

    f4 res;
    res.x = tanhf(acc.x);
    res.y = tanhf(acc.y);
    res.z = tanhf(acc.z);
    res.w = tanhf(acc.w);
    *(f4*)(out + (size_t)b * N_NODES + i0) = res;
  }
}

// ---------------------------------------------------------------------------
// Prep kernel: thr[j*N+i] = sigmoid(-log_alpha[j*N+i]); Wt[j*N+i] = W[i*N+j]
// 1 MB each into d_ws; removes transcendentals + uncoalesced W access from
// the hot loop. All 512 batch-WGs then re-read these 2 MB out of L2.
// sigmoid via v_rcp_f32 (threshold only needs ~1 ulp).
// ---------------------------------------------------------------------------
__global__ __launch_bounds__(256) void gumbel_prep_kernel(
    const float* __restrict__ log_alpha,
    const float* __restrict__ W,
    float* __restrict__ thr,
    float* __restrict__ Wt) {
  int idx = blockIdx.x * 256 + threadIdx.x;
  if (idx >= N_NODES * N_NODES) return;
  int j = idx / N_NODES;
  int i = idx - j * N_NODES;
  float la = log_alpha[idx];
  thr[idx] = __builtin_amdgcn_rcpf(1.0f + __expf(la));  // sigmoid(-la)
  Wt[idx]  = W[i * N_NODES + j];                        // transpose
}

extern "C" void kernel_launch(void* const* d_in, const int* in_sizes, int n_in,
                              void* d_out, int out_size, void* d_ws, size_t ws_size,
                              hipStream_t stream) {
  (void)in_sizes; (void)n_in; (void)out_size; (void)ws_size;

  const float* x         = (const float*)d_in[0];  // [B, N]
  const float* log_alpha = (const float*)d_in[1];  // [N, N]
  const float* W         = (const float*)d_in[2];  // [N, N] (torch [out,in])
  const float* u         = (const float*)d_in[3];  // [B, N, N]
  float* out = (float*)d_out;                      // [B, N]

  float* thr = (float*)d_ws;                       // [N, N] thresholds
  float* Wt  = thr + N_NODES * N_NODES;            // [N, N] W transposed

  const int prep_elems  = N_NODES * N_NODES;
  const int prep_blocks = (prep_elems + 255) / 256;
  gumbel_prep_kernel<<<prep_blocks, 256, 0, stream>>>(log_alpha, W, thr, Wt);

  gumbel_mlp_kernel<<<BATCH, 128, 0, stream>>>(x, u, thr, Wt, out);
}